// BERTEmbedding_40200893891003
// MI455X (gfx1250) — compile-verified
//
#include <hip/hip_runtime.h>
#include <hip/hip_bf16.h>
#include <stdint.h>

// Problem constants from the reference
#define B_DIM 16
#define V_DIM 4096
#define E_DIM 768
#define TPB   (E_DIM / 4)   // 192 threads: one float4 of E per thread (6 waves, wave32)

// -ln(10000)/E  (div_term[k] = exp(2k * (-ln(1e4)/E)) = exp(e_even * (-ln(1e4)/E)))
#define NEG_LOG1E4_OVER_E (-9.210340371976184e+0f / 768.0f)

// Native clang vector type: valid for __builtin_nontemporal_store and gives B128 ops
typedef float v4f __attribute__((ext_vector_type(4)));

__global__ __launch_bounds__(TPB)
void bert_embedding_kernel(const float* __restrict__ seq,
                           const float* __restrict__ flux_w,
                           const float* __restrict__ flux_b,
                           const float* __restrict__ time_w,
                           const float* __restrict__ time_b,
                           float* __restrict__ out)
{
    // LDS layout (floats): [0,768) fw row | [768,1536) fb | [1536,2304) tw |
    //                      [2304,3072) tb | [3072,3072+48) staged sequence scalars
    __shared__ float smem[4 * E_DIM + 64];

    const int tid = threadIdx.x;
    const int v   = blockIdx.x;

    // Low 32 bits of a flat LDS pointer == LDS byte offset
    const uint32_t lds_base = (uint32_t)(uintptr_t)(&smem[0]);
    const uint32_t lds_dst  = lds_base + (uint32_t)tid * 16u;

    const size_t row = (size_t)v * E_DIM;
    const uint64_t ga_fw = (uint64_t)(uintptr_t)(flux_w + row) + (uint64_t)tid * 16u;
    const uint64_t ga_fb = (uint64_t)(uintptr_t)(flux_b + row) + (uint64_t)tid * 16u;
    const uint64_t ga_tw = (uint64_t)(uintptr_t)(time_w + row) + (uint64_t)tid * 16u;
    const uint64_t ga_tb = (uint64_t)(uintptr_t)(time_b + row) + (uint64_t)tid * 16u;

    // gfx1250 async copies: global -> LDS, 16B per lane per op (tracked by ASYNCcnt).
    asm volatile("global_load_async_to_lds_b128 %0, %1, off"
                 :: "v"(lds_dst),                         "v"(ga_fw) : "memory");
    asm volatile("global_load_async_to_lds_b128 %0, %1, off"
                 :: "v"(lds_dst + 1u * E_DIM * 4u),       "v"(ga_fb) : "memory");
    asm volatile("global_load_async_to_lds_b128 %0, %1, off"
                 :: "v"(lds_dst + 2u * E_DIM * 4u),       "v"(ga_tw) : "memory");
    asm volatile("global_load_async_to_lds_b128 %0, %1, off"
                 :: "v"(lds_dst + 3u * E_DIM * 4u),       "v"(ga_tb) : "memory");

    // Stage the 16*3 sequence scalars for this v (layout [B, V, 3]).
    if (tid < B_DIM * 3) {
        const int b = tid / 3;
        const int c = tid - b * 3;
        smem[4 * E_DIM + tid] = seq[((size_t)b * V_DIM + v) * 3 + c];
    }

    // Wait for our async copies, then make LDS visible workgroup-wide.
    asm volatile("s_wait_asynccnt 0" ::: "memory");
    __syncthreads();

    // Pull this thread's 16 weight floats out of LDS once (reused 16x over b).
    const v4f wf = ((const v4f*)(&smem[0 * E_DIM]))[tid];
    const v4f bf = ((const v4f*)(&smem[1 * E_DIM]))[tid];
    const v4f wt = ((const v4f*)(&smem[2 * E_DIM]))[tid];
    const v4f bt = ((const v4f*)(&smem[3 * E_DIM]))[tid];

    // Sinusoidal terms for this thread's e-range: e0..e0+3, pairs (e0,e0+1),(e0+2,e0+3)
    const int   e0 = tid * 4;
    const float t0 = __expf((float)e0       * NEG_LOG1E4_OVER_E);
    const float t1 = __expf((float)(e0 + 2) * NEG_LOG1E4_OVER_E);

    #pragma unroll
    for (int b = 0; b < B_DIM; ++b) {
        const float s_flux = smem[4 * E_DIM + b * 3 + 0];
        const float s_pass = smem[4 * E_DIM + b * 3 + 1];
        const float s_time = smem[4 * E_DIM + b * 3 + 2];

        const float a0 = s_pass * t0;
        const float a1 = s_pass * t1;

        v4f o;
        o.x = fmaf(s_flux, wf.x, bf.x) + fmaf(s_time, wt.x, bt.x) + __sinf(a0);
        o.y = fmaf(s_flux, wf.y, bf.y) + fmaf(s_time, wt.y, bt.y) + __cosf(a0);
        o.z = fmaf(s_flux, wf.z, bf.z) + fmaf(s_time, wt.z, bt.z) + __sinf(a1);
        o.w = fmaf(s_flux, wf.w, bf.w) + fmaf(s_time, wt.w, bt.w) + __cosf(a1);

        // Write-once 201 MB stream: non-temporal so it doesn't thrash L2.
        v4f* dst = (v4f*)(out + ((size_t)b * V_DIM + v) * E_DIM);
        __builtin_nontemporal_store(o, &dst[tid]);
    }
}

extern "C" void kernel_launch(void* const* d_in, const int* in_sizes, int n_in,
                              void* d_out, int out_size, void* d_ws, size_t ws_size,
                              hipStream_t stream) {
    (void)in_sizes; (void)n_in; (void)d_ws; (void)ws_size; (void)out_size;

    const float* seq    = (const float*)d_in[0];   // [B, V, 3]
    const float* flux_w = (const float*)d_in[1];   // [V, E]
    const float* flux_b = (const float*)d_in[2];   // [V, E]
    const float* time_w = (const float*)d_in[3];   // [V, E]
    const float* time_b = (const float*)d_in[4];   // [V, E]
    float*       out    = (float*)d_out;           // [B, V, E]

    bert_embedding_kernel<<<dim3(V_DIM), dim3(TPB), 0, stream>>>(
        seq, flux_w, flux_b, time_w, time_b, out);
}